// edgeConv_73254962200982
// MI455X (gfx1250) — compile-verified
//
#include <hip/hip_runtime.h>

#define B_SZ  8
#define DIMS  64
#define NPTS  2048
#define FOUT  64
#define KNN   20
#define CIN   128          // 2*DIMS
#define BN_EPS 1e-5f

typedef __attribute__((ext_vector_type(2))) float v2f;
typedef __attribute__((ext_vector_type(8))) float v8f;

// ---------------- small utility kernels ----------------

__global__ void zero_stats_kernel(float* __restrict__ stats) {
  stats[threadIdx.x] = 0.0f;              // 128 threads: sum[64] + sumsq[64]
}

__global__ __launch_bounds__(256) void sqnorm_kernel(const float* __restrict__ x,
                                                     float* __restrict__ xs) {
  int idx = blockIdx.x * 256 + threadIdx.x;     // b*NPTS + n
  int b = idx >> 11;
  int n = idx & (NPTS - 1);
  const float* xb = x + ((size_t)b * DIMS) * NPTS + n;
  float s = 0.0f;
#pragma unroll 8
  for (int d = 0; d < DIMS; ++d) {
    float v = xb[(size_t)d * NPTS];
    s += v * v;
  }
  xs[idx] = s;
}

__global__ void finalize_stats_kernel(float* __restrict__ stats) {
  int o = threadIdx.x;                    // 64 threads
  const float cnt = (float)((size_t)B_SZ * NPTS * KNN);
  float m   = stats[o] / cnt;
  float var = stats[64 + o] / cnt - m * m;
  stats[128 + o] = m;
  stats[192 + o] = rsqrtf(var + BN_EPS);
}

// ---------------- fused dist-GEMM (WMMA f32) + top-k ----------------
// 1 workgroup = (batch b, 16 consecutive rows). 8 waves, each computes 16
// column tiles of 16x16 via chained V_WMMA_F32_16X16X4_F32, applies
// 2*sigmoid-1, stores 16 full rows (128 KB) in LDS, then 16 threads/row do
// 21 rounds of stable arg-min (round 0 = self, dropped).

__global__ __launch_bounds__(256) void knn_kernel(const float* __restrict__ x,
                                                  const float* __restrict__ xs,
                                                  int* __restrict__ idxK,
                                                  float* __restrict__ distK) {
  extern __shared__ char smem[];
  float* rowlds = (float*)smem;                 // [16][NPTS]
  const int tid  = threadIdx.x;
  const int lane = tid & 31;
  const int wave = tid >> 5;
  const int b = blockIdx.x >> 7;
  const int I = (blockIdx.x & 127) << 4;
  const float* xb  = x  + (size_t)b * DIMS * NPTS;
  const float* xsb = xs + b * NPTS;

  const int M     = lane & 15;                  // row (A) / col (B,C) within tile
  const int kb    = (lane >> 4) << 1;           // K-pair base: 0 or 2
  const int rhalf = (lane >> 4) << 3;           // C row offset: 0 or 8

  // A fragments for 16 K-steps (rows I..I+15, all 64 dims), f32 16x16x4 layout
  v2f a[16];
#pragma unroll
  for (int s = 0; s < 16; ++s) {
    int d = 4 * s + kb;
    a[s].x = xb[(size_t)d       * NPTS + I + M];
    a[s].y = xb[(size_t)(d + 1) * NPTS + I + M];
  }
  float xsrow[8];
#pragma unroll
  for (int v = 0; v < 8; ++v) xsrow[v] = xsb[I + v + rhalf];

  for (int t = 0; t < 16; ++t) {
    int J = (wave + (t << 3)) << 4;             // this wave's column tile
    v8f c8 = {};
#pragma unroll
    for (int s = 0; s < 16; ++s) {
      int d = 4 * s + kb;
      v2f bb;
      bb.x = xb[(size_t)d       * NPTS + J + M];
      bb.y = xb[(size_t)(d + 1) * NPTS + J + M];
      c8 = __builtin_amdgcn_wmma_f32_16x16x4_f32(false, a[s], false, bb,
                                                 (short)0, c8, false, false);
    }
    int col = J + M;
    float xsj = xsb[col];
#pragma unroll
    for (int v = 0; v < 8; ++v) {
      float dval = -2.0f * c8[v] + xsrow[v] + xsj;
      float tval = 2.0f / (1.0f + __expf(-dval)) - 1.0f;   // 2*sigmoid(d)-1
      rowlds[(v + rhalf) * NPTS + col] = tval;
    }
  }
  __syncthreads();

  // stable top-(k+1) selection: 16 threads per row (contiguous half-wave)
  const int row = tid >> 4;
  const int sub = tid & 15;
  float* rowbuf = rowlds + row * NPTS;
  const size_t obase = ((size_t)b * NPTS + I + row) * KNN;
  for (int it = 0; it <= KNN; ++it) {
    float bv = 1e30f;
    int   bi = 0;
    for (int jj = sub; jj < NPTS; jj += 16) {
      float v = rowbuf[jj];
      if (v < bv) { bv = v; bi = jj; }          // keeps smallest index on ties
    }
#pragma unroll
    for (int off = 8; off > 0; off >>= 1) {     // reduce within 16 lanes
      float ov = __shfl_xor(bv, off, 16);
      int   oi = __shfl_xor(bi, off, 16);
      if (ov < bv || (ov == bv && oi < bi)) { bv = ov; bi = oi; }
    }
    if (sub == 0) {
      rowbuf[bi] = 1e30f;                       // knock out the winner
      if (it >= 1) {                            // drop entry 0 (self)
        idxK [obase + it - 1] = bi;
        distK[obase + it - 1] = bv;
      }
    }
    __syncthreads();
  }
}

// ---------------- edge-conv GEMM (WMMA f32) : stats / output ----------------
// 1 workgroup = (batch b, 32 points). Per k: build E[128][32] in LDS, each of
// the 8 waves owns a fixed (16-row, 16-col) tile -> running max over k stays
// in registers. mode 0: accumulate per-channel sum/sumsq. mode 1: BN+ReLU+max.

__global__ __launch_bounds__(256) void conv_kernel(
    const float* __restrict__ x, const float* __restrict__ Wg,
    const float* __restrict__ bias, const float* __restrict__ gamma,
    const float* __restrict__ beta, const int* __restrict__ idxK,
    const float* __restrict__ distK, float* __restrict__ stats,
    float* __restrict__ out, int mode) {
  extern __shared__ char smem[];
  float* Wt    = (float*)smem;                  // [CIN][FOUT] transposed W
  float* E     = Wt + CIN * FOUT;               // [CIN][32]
  float* sstat = E + CIN * 32;                  // [128]

  const int tid  = threadIdx.x;
  const int lane = tid & 31;
  const int wave = tid >> 5;
  const int b  = blockIdx.x >> 6;
  const int n0 = (blockIdx.x & 63) << 5;
  const float* xb = x + (size_t)b * DIMS * NPTS;

  // W (row-major [FOUT][CIN]) -> LDS transposed (conflict-free A-frag reads)
  for (int i = tid; i < FOUT * CIN; i += 256) {
    int o = i >> 7;
    int c = i & (CIN - 1);
    Wt[c * FOUT + o] = Wg[i];
  }
  if (tid < 128) sstat[tid] = 0.0f;
  __syncthreads();

  const int M     = lane & 15;
  const int kb    = (lane >> 4) << 1;
  const int rhalf = (lane >> 4) << 3;
  const int r  = wave & 3;                      // row tile  -> o0
  const int ct = wave >> 2;                     // col tile  -> jb
  const int o0 = r << 4;
  const int jb = ct << 4;

  // hoisted A fragments (W tile) — invariant across the k loop
  v2f a[16];
#pragma unroll
  for (int s = 0; s < 16; ++s) {
    int d = 4 * s + kb;
    a[s].x = Wt[(d    ) * FOUT + o0 + M];
    a[s].y = Wt[(d + 1) * FOUT + o0 + M];
  }

  float bs[8], mmean[8], minv[8], mg[8], mb[8], mx[8], ls[8], lq[8];
#pragma unroll
  for (int v = 0; v < 8; ++v) { mx[v] = 0.0f; ls[v] = 0.0f; lq[v] = 0.0f; }
#pragma unroll
  for (int v = 0; v < 8; ++v) bs[v] = bias[o0 + v + rhalf];
  if (mode == 1) {
#pragma unroll
    for (int v = 0; v < 8; ++v) {
      int o = o0 + v + rhalf;
      mmean[v] = stats[128 + o];
      minv[v]  = stats[192 + o];
      mg[v]    = gamma[o];
      mb[v]    = beta[o];
    }
  }

  const int j = tid & 31;                       // E column this thread fills
  const int g = tid >> 5;
  const int n = n0 + j;
  const size_t nkbase = ((size_t)b * NPTS + n) * KNN;

  for (int kk = 0; kk < KNN; ++kk) {
    __syncthreads();
    // ---- build E[128][32]: central rows 0..63, weighted neighbors 64..127
    int   nbr = idxK [nkbase + kk];
    float w   = 1.0f - distK[nkbase + kk];
#pragma unroll
    for (int i = 0; i < 8; ++i) {
      int c = g + 8 * i;                        // 0..63
      E[c * 32 + j] = xb[(size_t)c * NPTS + n];
    }
#pragma unroll
    for (int i = 0; i < 8; ++i) {
      int c = g + 8 * i;
      E[(c + 64) * 32 + j] = w * xb[(size_t)c * NPTS + nbr];
    }
    __syncthreads();

    // ---- 16x16 output tile via 16 chained f32 WMMA (K=64 -> CIN=128? no:
    //      16 steps x K4 = 64 ... CIN=128 => 32 steps) — use 32 K-steps
    v8f c8 = {};
#pragma unroll
    for (int s = 0; s < 32; ++s) {
      int d = 4 * s + kb;
      v2f bb;
      bb.x = E[(d    ) * 32 + jb + M];
      bb.y = E[(d + 1) * 32 + jb + M];
      // A frag for steps 16..31 read directly from LDS (register budget)
      v2f af;
      if (s < 16) { af = a[s]; }
      else {
        af.x = Wt[(d    ) * FOUT + o0 + M];
        af.y = Wt[(d + 1) * FOUT + o0 + M];
      }
      c8 = __builtin_amdgcn_wmma_f32_16x16x4_f32(false, af, false, bb,
                                                 (short)0, c8, false, false);
    }

    if (mode == 0) {
#pragma unroll
      for (int v = 0; v < 8; ++v) {
        float y = c8[v] + bs[v];
        ls[v] += y;
        lq[v] += y * y;
      }
    } else {
#pragma unroll
      for (int v = 0; v < 8; ++v) {
        float y  = c8[v] + bs[v];
        float yn = mg[v] * (y - mmean[v]) * minv[v] + mb[v];
        yn = fmaxf(yn, 0.0f);
        mx[v] = fmaxf(mx[v], yn);
      }
    }
  }

  if (mode == 0) {
#pragma unroll
    for (int v = 0; v < 8; ++v) {
      int o = o0 + v + rhalf;
      atomicAdd(&sstat[o],      ls[v]);
      atomicAdd(&sstat[64 + o], lq[v]);
    }
    __syncthreads();
    if (tid < 128) atomicAdd(&stats[tid], sstat[tid]);
  } else {
    int nn = n0 + jb + M;
#pragma unroll
    for (int v = 0; v < 8; ++v) {
      int o = o0 + v + rhalf;
      out[((size_t)b * FOUT + o) * NPTS + nn] = mx[v];
    }
  }
}

// ---------------- launch ----------------

extern "C" void kernel_launch(void* const* d_in, const int* in_sizes, int n_in,
                              void* d_out, int out_size, void* d_ws, size_t ws_size,
                              hipStream_t stream) {
  const float* x     = (const float*)d_in[0];
  const float* Wg    = (const float*)d_in[1];
  const float* bias  = (const float*)d_in[2];
  const float* gamma = (const float*)d_in[3];
  const float* beta  = (const float*)d_in[4];
  float* out = (float*)d_out;

  char* ws = (char*)d_ws;
  const size_t OFF_XS  = 0;                                   // 65536 B
  const size_t OFF_IDX = OFF_XS  + (size_t)B_SZ * NPTS * 4;   // 1.31 MB
  const size_t OFF_DK  = OFF_IDX + (size_t)B_SZ * NPTS * KNN * 4;
  const size_t OFF_ST  = OFF_DK  + (size_t)B_SZ * NPTS * KNN * 4;
  float* xs    = (float*)(ws + OFF_XS);
  int*   idxK  = (int*)  (ws + OFF_IDX);
  float* distK = (float*)(ws + OFF_DK);
  float* stats = (float*)(ws + OFF_ST);   // sum[64] sumsq[64] mean[64] invstd[64]

  const size_t knnLds  = (size_t)16 * NPTS * 4;                    // 128 KB
  const size_t convLds = (size_t)(CIN * FOUT + CIN * 32 + 128) * 4; // ~48.5 KB

  zero_stats_kernel<<<1, 128, 0, stream>>>(stats);
  sqnorm_kernel<<<(B_SZ * NPTS) / 256, 256, 0, stream>>>(x, xs);
  knn_kernel<<<B_SZ * (NPTS / 16), 256, knnLds, stream>>>(x, xs, idxK, distK);
  conv_kernel<<<B_SZ * (NPTS / 32), 256, convLds, stream>>>(
      x, Wg, bias, gamma, beta, idxK, distK, stats, out, 0);
  finalize_stats_kernel<<<1, 64, 0, stream>>>(stats);
  conv_kernel<<<B_SZ * (NPTS / 32), 256, convLds, stream>>>(
      x, Wg, bias, gamma, beta, idxK, distK, stats, out, 1);
}